// LocallyConnected2d_41094247088738
// MI455X (gfx1250) — compile-verified
//
#include <hip/hip_runtime.h>

typedef float v2f __attribute__((ext_vector_type(2)));
typedef float v8f __attribute__((ext_vector_type(8)));

#define AS1 __attribute__((address_space(1)))
#define AS3 __attribute__((address_space(3)))

#if defined(__has_builtin)
#  if __has_builtin(__builtin_amdgcn_global_load_async_to_lds_b32)
#    define HAVE_ASYNC_LDS 1
#  endif
#endif
#ifndef HAVE_ASYNC_LDS
#  define HAVE_ASYNC_LDS 0
#endif

__device__ __forceinline__ void async_drain() {
#if defined(__has_builtin)
#  if __has_builtin(__builtin_amdgcn_s_wait_asynccnt)
  __builtin_amdgcn_s_wait_asynccnt(0);
  return;
#  endif
#endif
  asm volatile("s_wait_asynccnt 0x0" ::: "memory");
}

// Problem dims (fixed by reference)
constexpr int Bn = 4, Cc = 64, On = 64, Ho = 32, Wo = 32, Kk = 3, Hi = 34;
constexpr int TAPS     = Kk * Kk;          // 9
constexpr int CK       = Cc * TAPS;        // 576 = reduction length (c,k)
constexpr int CPC      = 8;                // channels per staged chunk
constexpr int CHUNK_CK = CPC * TAPS;       // 72 ck per chunk
constexpr int NCHUNK   = Cc / CPC;         // 8
constexpr int LDSW     = On + 1;           // 65: pad row -> conflict-free column reads

constexpr int W_OSTRIDE = Cc * Ho * Wo * TAPS; // 589824
constexpr int W_CSTRIDE = Ho * Wo * TAPS;      // 9216
constexpr int W_HSTRIDE = Wo * TAPS;           // 288

__global__ __launch_bounds__(128, 1)
void lc2d_wmma_f32(const float* __restrict__ x,
                   const float* __restrict__ wgt,
                   float* __restrict__ out) {
  // [buf][ck_local][o] : transposed weight slab, padded  (2*72*65*4 = 37.4 KB)
  __shared__ float wlds[2][CHUNK_CK][LDSW];
  // [ck][b] : input patches for this (h,w)               (9 KB)
  __shared__ float plds[CK][4];

  const int t    = threadIdx.x;
  const int lane = t & 31;          // wave32
  const int wave = t >> 5;          // 0..3 -> o-group of 16
  const int hw   = blockIdx.x;      // 0..1023 spatial position
  const int h    = hw >> 5;
  const int w    = hw & 31;

  const float* wbase = wgt + h * W_HSTRIDE + w * TAPS;

  // ---- weight staging: thread owns o = t>>1 and 4 channels (t&1 half) ----
  const int so    = t >> 1;
  const int chalf = (t & 1) * 4;

#if HAVE_ASYNC_LDS
  // Async gather/scatter: global [o][c][k] -> LDS [ck][o], transposed in flight.
  // Each instruction is wave-wide (128 B), tracked on ASYNCcnt.
  auto stage_async = [&](int chunk, int buf) {
    const float* wp = wbase + so * W_OSTRIDE + chunk * CPC * W_CSTRIDE;
    #pragma unroll
    for (int j = 0; j < 4; ++j) {
      const float* g = wp + (chalf + j) * W_CSTRIDE;
      float*       l = &wlds[buf][(chalf + j) * TAPS][so];
      #pragma unroll
      for (int k = 0; k < TAPS; ++k)
        __builtin_amdgcn_global_load_async_to_lds_b32(
            (AS1 int*)(g + k), (AS3 int*)(l + k * LDSW), 0, 0);
    }
  };
#else
  float reg[4 * TAPS];
  auto stage = [&](int chunk) {
    const float* wp = wbase + so * W_OSTRIDE + chunk * CPC * W_CSTRIDE;
    #pragma unroll
    for (int j = 0; j < 4; ++j)
      #pragma unroll
      for (int k = 0; k < TAPS; ++k)
        reg[j * TAPS + k] = wp[(chalf + j) * W_CSTRIDE + k];
  };
  auto commit = [&](int buf) {
    #pragma unroll
    for (int j = 0; j < 4; ++j)
      #pragma unroll
      for (int k = 0; k < TAPS; ++k)
        wlds[buf][(chalf + j) * TAPS + k][so] = reg[j * TAPS + k];
  };
#endif

  // ---- WMMA setup: D(16x16) = A(16x4) * B(4x16) + C over K=576 ----
  v8f acc = {};
  const int hi = lane >> 4;   // lane half selects K {0,1} vs {2,3}
  const int n  = lane & 15;   // tile column = o within group
  const int bb = lane & 3;    // batch row for A (rows 4..15 replicate row b&3)
  const int og = wave * 16;

  auto compute = [&](int ch, int buf) {
    #pragma unroll
    for (int kk = 0; kk < CHUNK_CK / 4; ++kk) {
      const int ckl = kk * 4 + hi * 2;       // K layout: v0={K0,K2}, v1={K1,K3}
      const int ckg = ch * CHUNK_CK + ckl;
      v2f a, bf;
      a.x  = plds[ckg + 0][bb];
      a.y  = plds[ckg + 1][bb];
      bf.x = wlds[buf][ckl + 0][og + n];
      bf.y = wlds[buf][ckl + 1][og + n];
      acc = __builtin_amdgcn_wmma_f32_16x16x4_f32(
                false, a, false, bf, (short)0, acc, false, false);
    }
  };

#if HAVE_ASYNC_LDS
  stage_async(0, 0);
#else
  stage(0);
#endif

  // ---- stage input patches A[ck][b] (tiny, once per block) ----
  for (int idx = t; idx < CK * Bn; idx += 128) {
    int ck = idx >> 2;
    int b  = idx & 3;
    int c  = ck / TAPS;
    int k  = ck - c * TAPS;
    int kh = k / Kk;
    int kw = k - kh * Kk;
    plds[ck][b] = x[((b * Cc + c) * Hi + (h + kh)) * Hi + (w + kw)];
  }

#if HAVE_ASYNC_LDS
  async_drain();
  __syncthreads();
  for (int ch = 0; ch < NCHUNK; ++ch) {
    const int buf = ch & 1;
    if (ch + 1 < NCHUNK) stage_async(ch + 1, buf ^ 1);  // overlap with WMMA
    compute(ch, buf);
    if (ch + 1 < NCHUNK) async_drain();
    __syncthreads();
  }
#else
  commit(0);
  __syncthreads();
  for (int ch = 0; ch < NCHUNK; ++ch) {
    const int buf = ch & 1;
    if (ch + 1 < NCHUNK) stage(ch + 1);
    compute(ch, buf);
    if (ch + 1 < NCHUNK) commit(buf ^ 1);
    __syncthreads();
  }
#endif

  // ---- store: C/D layout -> VGPR r holds M=r (lanes 0..15); batches are M=0..3
  if (lane < 16) {
    const int o = og + n;
    #pragma unroll
    for (int r = 0; r < Bn; ++r)
      out[((r * On + o) * Ho + h) * Wo + w] = acc[r];
  }
}

extern "C" void kernel_launch(void* const* d_in, const int* /*in_sizes*/, int /*n_in*/,
                              void* d_out, int /*out_size*/, void* /*d_ws*/, size_t /*ws_size*/,
                              hipStream_t stream) {
  const float* x   = (const float*)d_in[0];   // (4,64,34,34) fp32
  const float* wgt = (const float*)d_in[1];   // (1,64,64,32,32,9) fp32
  float* out = (float*)d_out;                 // (4,64,32,32) fp32
  lc2d_wmma_f32<<<Ho * Wo, 128, 0, stream>>>(x, wgt, out);
}